// CFNet_49830210568878
// MI455X (gfx1250) — compile-verified
//
#include <hip/hip_runtime.h>
#include <stdint.h>

// ---------------- problem constants (shapes fixed by the harness) ----------
#define NB   8      // batch
#define CC   3      // channels
#define H2   512    // hi-res feature H (2H)
#define W2   512    // hi-res feature W (2W)
#define HH   256    // output H
#define WW   256    // output W
#define HW   (HH*WW)
#define KCH  100    // 4*ksize^2
#define NROW 6      // feat rows read per output row: 2h-2 .. 2h+3  (dy+beta<=5)
#define PW   520    // padded LDS row width in words: [4 align/pad][512][4 pad]
#define LDS_FLOATS (CC*NROW*PW)     // 9360 floats = 37440 B

// Derivation (verified): out[n,c2,h,w] = sum_t kern[n,(t%10)*10+t/10,h,w] *
//   feat[n, c, clamp(2h+dy+beta-2), clamp(4*(w-128*dy)+u+alpha-2)]
// with t = q*25 + alpha*5 + beta, k = c2*4+q, u = k/3, c = k%3,
// dy = (w>=128).  For fixed (q,beta), alpha=0..4 -> 5 consecutive columns.

// ---------------- CDNA5 async global->LDS plumbing -------------------------
typedef int   v4i __attribute__((ext_vector_type(4)));
typedef float v2f __attribute__((ext_vector_type(2)));
typedef __attribute__((address_space(1))) void glb_void_t;
typedef __attribute__((address_space(3))) void lds_void_t;
typedef __attribute__((address_space(1))) v4i  glb_v4i_t;
typedef __attribute__((address_space(3))) v4i  lds_v4i_t;

#if __has_builtin(__builtin_amdgcn_global_load_async_to_lds_b128)
#define FAC_ASYNC 1
#else
#define FAC_ASYNC 0
#endif

__device__ __forceinline__ void stage16(const float* gp, float* lp) {
#if FAC_ASYNC
    glb_v4i_t* gsrc = (glb_v4i_t*)(glb_void_t*)(void*)gp;
    lds_v4i_t* ldst = (lds_v4i_t*)(lds_void_t*)(void*)lp;
    __builtin_amdgcn_global_load_async_to_lds_b128(gsrc, ldst, /*offset=*/0, /*cpol=*/0);
#else
    *(float4*)lp = *(const float4*)gp;
#endif
}

__device__ __forceinline__ void wait_async_done() {
#if FAC_ASYNC
#if __has_builtin(__builtin_amdgcn_s_wait_asynccnt)
    __builtin_amdgcn_s_wait_asynccnt(0);
#else
    asm volatile("s_wait_asynccnt 0" ::: "memory");
#endif
#endif
}

__device__ __forceinline__ v2f pk_fma(v2f a, v2f b, v2f c) {
#if __has_builtin(__builtin_elementwise_fma)
    return __builtin_elementwise_fma(a, b, c);   // -> v_pk_fma_f32
#else
    return a * b + c;                             // contracted fma pair
#endif
}

// ---------------- kernel ---------------------------------------------------
__global__ __launch_bounds__(256)
void fac_kernel(const float* __restrict__ feat,    // (N,3,512,512)
                const float* __restrict__ kern,    // (N,100,256,256)
                float* __restrict__ out) {         // (N,3,256,256)
    __shared__ __align__(16) float lds[LDS_FLOATS];

    const int w = threadIdx.x;    // 0..255 (output column)
    const int h = blockIdx.x;     // 0..255
    const int n = blockIdx.y;     // 0..7

    const float* featn = feat + (size_t)n * CC * H2 * W2;

    // ---- async-stage 3 ch x 6 edge-clamped rows (512 cols) into LDS -------
    // 18 rows x 128 chunks = 2304 = 9 * 256 : exactly 9 b128 copies/thread.
    for (int it = 0; it < 9; ++it) {
        int idx   = it * 256 + w;
        int rid   = idx >> 7;               // 0..17
        int chunk = idx & 127;              // 16B chunk in row
        int c = rid / NROW;
        int r = rid - c * NROW;
        int srow = 2 * h - 2 + r;           // replicate-pad clamp
        srow = srow < 0 ? 0 : (srow > H2 - 1 ? H2 - 1 : srow);
        const float* gp = featn + ((size_t)c * H2 + srow) * W2 + chunk * 4;
        float*       lp = &lds[(c * NROW + r) * PW + 4 + chunk * 4]; // 16B aligned
        stage16(gp, lp);
    }
    // ---- replicated edge columns (cols -2,-1 and 512,513) -----------------
    if (w < CC * NROW) {
        int c = w / NROW;
        int r = w - c * NROW;
        int srow = 2 * h - 2 + r;
        srow = srow < 0 ? 0 : (srow > H2 - 1 ? H2 - 1 : srow);
        const float* grow = featn + ((size_t)c * H2 + srow) * W2;
        float vl = grow[0];
        float vr = grow[W2 - 1];
        int B = (c * NROW + r) * PW;
        lds[B + 2]   = vl;  lds[B + 3]   = vl;   // cols -2, -1
        lds[B + 516] = vr;  lds[B + 517] = vr;   // cols 512, 513
    }
    wait_async_done();
    __syncthreads();

    // ---- fully-unrolled 100-tap dot product, 3 channels -------------------
    const float* kb = kern + (size_t)n * KCH * HW + (size_t)h * WW + w;
    const int dy = w >> 7;          // 0 or 1
    const int wl = w & 127;         // column within half

    // 15 row base pointers: word(col x) = row*PW + 4 + x ; group starts at
    // col = 4*wl - 2 + u, so base points at col "4*wl - 2" (word 4*wl + 2).
    const float* pc[CC][5];
#pragma unroll
    for (int c = 0; c < CC; ++c)
#pragma unroll
        for (int b = 0; b < 5; ++b)
            pc[c][b] = &lds[(c * NROW + dy + b) * PW + 4 * wl + 2];

    v2f   acc2[CC] = {{0.f, 0.f}, {0.f, 0.f}, {0.f, 0.f}};
    float acc1[CC] = {0.f, 0.f, 0.f};

#pragma unroll
    for (int q = 0; q < 4; ++q) {
#pragma unroll
        for (int b = 0; b < 5; ++b) {
            // 5 taps t = q*25 + a*5 + b  (a = column tap)
            float kv[5];
#pragma unroll
            for (int a = 0; a < 5; ++a) {
                const int t  = q * 25 + a * 5 + b;
                const int ch = (t % 10) * 10 + t / 10;
                // 210 MB streamed once -> non-temporal, keep L2 for feat
                kv[a] = __builtin_nontemporal_load(kb + (size_t)ch * HW);
            }
            v2f kv01 = {kv[0], kv[1]};
            v2f kv23 = {kv[2], kv[3]};
#pragma unroll
            for (int c2 = 0; c2 < CC; ++c2) {
                const int k = c2 * 4 + q;
                const int u = k / 3;          // 0..3
                const int c = k % 3;          // feat channel
                const float* p = pc[c][b] + u;
                v2f f01 = {p[0], p[1]};
                v2f f23 = {p[2], p[3]};
                acc2[c2] = pk_fma(kv01, f01, acc2[c2]);
                acc2[c2] = pk_fma(kv23, f23, acc2[c2]);
                acc1[c2] = fmaf(kv[4], p[4], acc1[c2]);
            }
        }
    }

    float* o = out + ((size_t)(n * CC) * HH + h) * WW + w;
#pragma unroll
    for (int c2 = 0; c2 < CC; ++c2)
        o[(size_t)c2 * HW] = acc2[c2].x + acc2[c2].y + acc1[c2];
}

// ---------------- harness entry point --------------------------------------
extern "C" void kernel_launch(void* const* d_in, const int* in_sizes, int n_in,
                              void* d_out, int out_size, void* d_ws, size_t ws_size,
                              hipStream_t stream) {
    const float* feat = (const float*)d_in[0];   // (8,3,512,512) f32
    const float* kern = (const float*)d_in[1];   // (8,100,256,256) f32
    // d_in[2] = ksize (==5), baked in at compile time
    float* out = (float*)d_out;                  // (8,3,256,256) f32

    dim3 grid(HH, NB);
    dim3 block(WW);
    fac_kernel<<<grid, block, 0, stream>>>(feat, kern, out);
}